// HierarchyPreservationLoss_13426067767716
// MI455X (gfx1250) — compile-verified
//
#include <hip/hip_runtime.h>
#include <math.h>

// Problem constants from the reference
#define N_PTS   8192
#define DIM     64
#define MM      2048
#define LDE     65                 // embeddings row stride (x0 | xs[64])
#define NT      (N_PTS / 16)       // 512 tile rows/cols
#define NTILES  ((NT * (NT + 1)) / 2)  // 131328 upper-tri tiles (incl. diagonal)

typedef __attribute__((ext_vector_type(2))) float v2f;
typedef __attribute__((ext_vector_type(8))) float v8f;

__global__ void hp_init_ws(float* ws) {
    if (threadIdx.x < 8) ws[threadIdx.x] = 0.0f;
}

// One wave computes one 16x16 tile of the Minkowski Gram matrix with
// V_WMMA_F32_16X16X4_F32 (17 steps: 16 for xs.xs^T, 1 folding -x0*x0^T),
// then applies mask/arccosh/gather branchlessly and accumulates 6 sums.
__global__ __launch_bounds__(256) void hp_main(
    const float* __restrict__ emb,    // (8192, 65) row-major: [x0, xs0..xs63]
    const float* __restrict__ tdist,  // (2048, 2048)
    const int*   __restrict__ gidx,   // (8192)
    float* __restrict__ gsum)         // 6 global accumulators in d_ws
{
    const int lane  = threadIdx.x & 31;
    const int l15   = lane & 15;     // M (for A) / N (for B,C) within tile
    const int lhalf = lane >> 4;     // selects K-pair 0..1 vs 2..3 (A/B), M+8 (C)

    const int      wavesPerBlock = blockDim.x >> 5;
    const unsigned waveId = blockIdx.x * wavesPerBlock + (threadIdx.x >> 5);
    const unsigned nWaves = gridDim.x * wavesPerBlock;

    float cnt = 0.f, se = 0.f, sg = 0.f, see = 0.f, sgg = 0.f, seg = 0.f;

    for (unsigned t = waveId; t < NTILES; t += nWaves) {
        // Decode upper-triangular tile (r, c), r <= c. Uniform across the wave,
        // so EXEC stays all-ones for the WMMA block (ISA requirement). The raw
        // v_sqrt_f32 estimate is corrected by the integer loops below.
        const float bb = 2.0f * (float)NT + 1.0f;
        unsigned r = (unsigned)((bb - __builtin_amdgcn_sqrtf(
                                          bb * bb - 8.0f * (float)t)) * 0.5f);
        while ((r + 1u) * NT - ((r + 1u) * r) / 2u <= t) ++r;      // fix undershoot
        while (r * NT - (r * (r - 1u)) / 2u > t)       --r;        // fix overshoot
        const unsigned c = r + (t - (r * NT - (r * (r - 1u)) / 2u));

        const int iBase = (int)r * 16;
        const int jBase = (int)c * 16;
        const float* aRow = emb + (iBase + l15) * LDE;  // A: M = l15, rows iBase..
        const float* bRow = emb + (jBase + l15) * LDE;  // B: N = l15, rows jBase..

        v8f acc = {0.f, 0.f, 0.f, 0.f, 0.f, 0.f, 0.f, 0.f};
        const int kOff = lhalf * 2;  // lanes 16-31 hold K=2,3 per f32 A/B layout

        #pragma unroll
        for (int kc = 0; kc < 16; ++kc) {
            const int k = kc * 4 + kOff;
            v2f a, b;
            a.x = aRow[1 + k];  a.y = aRow[2 + k];   // xs starts at col 1
            b.x = bRow[1 + k];  b.y = bRow[2 + k];
            acc = __builtin_amdgcn_wmma_f32_16x16x4_f32(
                false, a, false, b, (short)0, acc, false, false);
        }
        // Fold the rank-1 Minkowski correction: inner -= x0_i * x0_j.
        // Only K=0 (VGPR0, lanes 0-15) carries data; the rest is zero.
        {
            v2f a, b;
            a.x = (lhalf == 0) ? -aRow[0] : 0.0f;  a.y = 0.0f;
            b.x = (lhalf == 0) ?  bRow[0] : 0.0f;  b.y = 0.0f;
            acc = __builtin_amdgcn_wmma_f32_16x16x4_f32(
                false, a, false, b, (short)0, acc, false, false);
        }

        // C layout: VGPR rr, lane l -> M = rr + 8*lhalf, N = l15.
        const int Nj = jBase + l15;
        const int gj = gidx[Nj];

        // Row gt-indices: 8 contiguous ints at 32B-aligned address -> 2x int4.
        const int4 giA = *(const int4*)(gidx + iBase + lhalf * 8);
        const int4 giB = *(const int4*)(gidx + iBase + lhalf * 8 + 4);
        const int girow[8] = {giA.x, giA.y, giA.z, giA.w,
                              giB.x, giB.y, giB.z, giB.w};

        // Phase 1: issue all 8 L2 gathers so they clause together.
        float g[8];
        #pragma unroll
        for (int rr = 0; rr < 8; ++rr)
            g[rr] = tdist[girow[rr] * MM + gj];

        // Phase 2: branchless masked accumulation (mask true ~99% of the time,
        // so exec-mask branching would only add saveexec/branch overhead).
        // arccosh arguments are in the safe range (x >= 1+1e-7 so x^2-1 >= 2e-7,
        // log arg >= 1): use raw v_sqrt_f32 / v_log_f32 with no OCML fixups.
        #pragma unroll
        for (int rr = 0; rr < 8; ++rr) {
            const int   Mi = iBase + rr + lhalf * 8;
            const float m  = ((Mi < Nj) && (g[rr] >= 0.1f)) ? 1.0f : 0.0f;
            const float x  = fmaxf(-acc[rr], 1.0f + 1e-7f);
            const float e  = __builtin_amdgcn_logf(
                x + __builtin_amdgcn_sqrtf(fmaf(x, x, -1.0f)));  // arccosh
            const float em_ = e * m;
            const float gm_ = g[rr] * m;
            cnt += m;
            se  += em_;
            sg  += gm_;
            see  = fmaf(em_, e,     see);   // m*e^2 (m is 0/1)
            sgg  = fmaf(gm_, g[rr], sgg);   // m*g^2
            seg  = fmaf(em_, g[rr], seg);   // m*e*g
        }
    }

    // Block reduction via LDS float atomics, then one global atomic per slot.
    __shared__ float red[6];
    if (threadIdx.x < 6) red[threadIdx.x] = 0.0f;
    __syncthreads();
    atomicAdd(&red[0], cnt);
    atomicAdd(&red[1], se);
    atomicAdd(&red[2], sg);
    atomicAdd(&red[3], see);
    atomicAdd(&red[4], sgg);
    atomicAdd(&red[5], seg);
    __syncthreads();
    if (threadIdx.x < 6) atomicAdd(&gsum[threadIdx.x], red[threadIdx.x]);
}

__global__ void hp_finalize(const float* __restrict__ gsum, float* __restrict__ out) {
    const float cnt = gsum[0];
    const float se  = gsum[1], sg  = gsum[2];
    const float see = gsum[3], sgg = gsum[4], seg = gsum[5];
    const float em = se / cnt + 1e-8f;   // emb_mean + EPS
    const float gm = sg / cnt + 1e-8f;   // gt_mean + EPS
    // sum(diff^2 * m) = See/em^2 - 2*Seg/(em*gm) + Sgg/gm^2
    const float mse = (see / (em * em) - 2.0f * seg / (em * gm) + sgg / (gm * gm)) / cnt;
    out[0] = 0.1f * mse;                 // WEIGHT * mse
}

extern "C" void kernel_launch(void* const* d_in, const int* in_sizes, int n_in,
                              void* d_out, int out_size, void* d_ws, size_t ws_size,
                              hipStream_t stream) {
    const float* emb   = (const float*)d_in[0];  // embeddings (8192, 65) f32
    const float* tdist = (const float*)d_in[1];  // tree_distances (2048, 2048) f32
    const int*   gidx  = (const int*)d_in[2];    // gt_indices (8192) i32
    float* gsum = (float*)d_ws;

    hp_init_ws<<<1, 64, 0, stream>>>(gsum);
    hp_main<<<1024, 256, 0, stream>>>(emb, tdist, gidx, gsum);
    hp_finalize<<<1, 1, 0, stream>>>(gsum, (float*)d_out);
}